// DualGATConv_48017734369677
// MI455X (gfx1250) — compile-verified
//
#include <hip/hip_runtime.h>
#include <math.h>

#define H_HEADS 4
#define C_CH 32
#define F_DIM 128
#define NEG_SLOPE 0.2f

typedef __attribute__((ext_vector_type(2))) float v2f;
typedef __attribute__((ext_vector_type(8))) float v8f;

// ---------------- fill ----------------
__global__ void fill_kernel(float* __restrict__ p, long n, float v) {
    long i = (long)blockIdx.x * blockDim.x + threadIdx.x;
    long stride = (long)gridDim.x * blockDim.x;
    for (; i < n; i += stride) p[i] = v;
}

// ---------------- GEMM + fused attention coefficients ----------------
// Hout[rows,128] = X[rows,128] @ W[128,128]   (WMMA f32 16x16x4)
// as[r,h] = sum_c Hout[r, h*32+c] * a_src[h*32+c]   (ad likewise)
//
// Block = 256 threads = 8 waves; each wave owns a 16-row stripe and all 8
// column tiles. W is cached in LDS pair-packed: element (k,col) lives at
//   Wlds[(k>>1)*256 + csw*2 + (k&1)],  csw = (col + 16*((k>>1)&1)) & 127
// so each lane's B operand is one aligned ds_load_b64, and the 16-column
// rotation on odd k-pairs puts the two wave halves on disjoint LDS bank sets.
__global__ __launch_bounds__(256)
void gemm_wmma_kernel(const float* __restrict__ X, const float* __restrict__ W,
                      const float* __restrict__ a_src, const float* __restrict__ a_dst,
                      float* __restrict__ Hout, float* __restrict__ as,
                      float* __restrict__ ad, int rows) {
    extern __shared__ float Wlds[];                    // 128*128 floats = 64 KB
    for (int i = threadIdx.x; i < (F_DIM * F_DIM) / 2; i += blockDim.x) {
        int kp  = i >> 7;                              // k-pair row: 0..63
        int col = i & 127;
        int cs  = (col + ((kp & 1) << 4)) & 127;       // bank-deconflict rotation
        float2 w2;
        w2.x = W[(size_t)(2 * kp)     * F_DIM + col];
        w2.y = W[(size_t)(2 * kp + 1) * F_DIM + col];
        *(float2*)(Wlds + (size_t)kp * 256 + cs * 2) = w2;
    }
    __syncthreads();

    const int lane = threadIdx.x & 31;
    const int wave = threadIdx.x >> 5;
    const int lm   = lane & 15;
    const int lh   = lane >> 4;                        // 0 or 1 (K half / row half)
    const int rowBase = blockIdx.x * 128 + wave * 16;
    if (rowBase >= rows) return;                       // uniform per wave

    int aRow = rowBase + lm;
    if (aRow >= rows) aRow = rows - 1;                 // clamp loads; stores guarded

    v8f acc[8];
    v8f zero = {0.f, 0.f, 0.f, 0.f, 0.f, 0.f, 0.f, 0.f};
#pragma unroll
    for (int ct = 0; ct < 8; ++ct) acc[ct] = zero;

    const float* xrow = X + (size_t)aRow * F_DIM;
    for (int k0 = 0; k0 < F_DIM; k0 += 4) {
        const int kp = (k0 >> 1) + lh;                 // k0/2 even => kp parity == lh
        v2f a;
        a.x = xrow[2 * kp];
        a.y = xrow[2 * kp + 1];
        const float* bbase = Wlds + (size_t)kp * 256;
#pragma unroll
        for (int ct = 0; ct < 8; ++ct) {
            const int cs = (ct * 16 + lm + 16 * lh) & 127;   // static per lane/ct
            v2f b = *(const v2f*)(bbase + cs * 2);
            acc[ct] = __builtin_amdgcn_wmma_f32_16x16x4_f32(
                false, a, false, b, (short)0, acc[ct], false, false);
        }
    }

    // ---- store H tile
#pragma unroll
    for (int ct = 0; ct < 8; ++ct) {
#pragma unroll
        for (int g = 0; g < 8; ++g) {
            int r = rowBase + g + 8 * lh;
            if (r < rows) Hout[(size_t)r * F_DIM + ct * 16 + lm] = acc[ct][g];
        }
    }

    // ---- fused per-head attention coefficients (head h = column tiles 2h, 2h+1)
    float asv[8], adv[8];
#pragma unroll
    for (int ct = 0; ct < 8; ++ct) {
        asv[ct] = a_src[ct * 16 + lm];
        adv[ct] = a_dst[ct * 16 + lm];
    }
#pragma unroll
    for (int h = 0; h < 4; ++h) {
#pragma unroll
        for (int g = 0; g < 8; ++g) {
            float ss = acc[2 * h][g] * asv[2 * h];
            ss = fmaf(acc[2 * h + 1][g], asv[2 * h + 1], ss);
            float sd = acc[2 * h][g] * adv[2 * h];
            sd = fmaf(acc[2 * h + 1][g], adv[2 * h + 1], sd);
#pragma unroll
            for (int off = 8; off > 0; off >>= 1) {    // reduce within each 16-lane half
                ss += __shfl_xor(ss, off, 32);
                sd += __shfl_xor(sd, off, 32);
            }
            int r = rowBase + g + 8 * lh;
            if (lm == 0 && r < rows) {
                as[(size_t)r * H_HEADS + h] = ss;
                ad[(size_t)r * H_HEADS + h] = sd;
            }
        }
    }
}

// ---------------- float atomic max (int max / uint min trick) ----------------
__device__ inline void atomicMaxFloat(float* addr, float val) {
    if (val >= 0.f)
        atomicMax((int*)addr, __float_as_int(val));
    else
        atomicMin((unsigned int*)addr, __float_as_uint(val));
}

// ---------------- edge scores + segment max ----------------
__global__ void score_kernel(const float* __restrict__ as, const float* __restrict__ ad,
                             const int* __restrict__ ei, int E, int Nn,
                             float* __restrict__ alpha, float* __restrict__ m) {
    int id = blockIdx.x * blockDim.x + threadIdx.x;
    int total = E + Nn;
    if (id >= total) return;
    int src = (id < E) ? ei[id]     : id - E;
    int dst = (id < E) ? ei[E + id] : id - E;
    float4 s = *(const float4*)(as + (size_t)src * 4);
    float4 d = *(const float4*)(ad + (size_t)dst * 4);
    float ev[4] = {s.x + d.x, s.y + d.y, s.z + d.z, s.w + d.w};
#pragma unroll
    for (int h = 0; h < 4; ++h) {
        float e = ev[h];
        e = (e > 0.f) ? e : NEG_SLOPE * e;
        ev[h] = e;
        atomicMaxFloat(m + (size_t)dst * 4 + h, e);
    }
    *(float4*)(alpha + (size_t)id * 4) = make_float4(ev[0], ev[1], ev[2], ev[3]);
}

// ---------------- exp(e - m[dst]) + segment sum ----------------
__global__ void expsum_kernel(const int* __restrict__ ei, int E, int Nn,
                              const float* __restrict__ m, float* __restrict__ alpha,
                              float* __restrict__ z) {
    int id = blockIdx.x * blockDim.x + threadIdx.x;
    int total = E + Nn;
    if (id >= total) return;
    int dst = (id < E) ? ei[E + id] : id - E;
    float4 e  = *(const float4*)(alpha + (size_t)id * 4);
    float4 mv = *(const float4*)(m + (size_t)dst * 4);
    float4 p;
    p.x = __expf(e.x - mv.x);
    p.y = __expf(e.y - mv.y);
    p.z = __expf(e.z - mv.z);
    p.w = __expf(e.w - mv.w);
    *(float4*)(alpha + (size_t)id * 4) = p;
    atomicAdd(&z[(size_t)dst * 4 + 0], p.x);
    atomicAdd(&z[(size_t)dst * 4 + 1], p.y);
    atomicAdd(&z[(size_t)dst * 4 + 2], p.z);
    atomicAdd(&z[(size_t)dst * 4 + 3], p.w);
}

// ---------------- normalize alpha + scatter-aggregate messages ----------------
// One wave per edge. Lane covers 4 channels (one head): acc[dst] += alpha*h[src].
__global__ void agg_kernel(const float* __restrict__ hbuf, const float* __restrict__ z,
                           const int* __restrict__ ei, int E, int Nn,
                           float* __restrict__ alpha, float* __restrict__ acc) {
    int widx = blockIdx.x * (blockDim.x >> 5) + (threadIdx.x >> 5);
    int lane = threadIdx.x & 31;
    int total = E + Nn;
    if (widx >= total) return;                          // uniform per wave
    int src = (widx < E) ? ei[widx]     : widx - E;
    int dst = (widx < E) ? ei[E + widx] : widx - E;
    int head = lane >> 3;
    float p  = alpha[(size_t)widx * 4 + head];
    float zz = z[(size_t)dst * 4 + head];
    float al = p / zz;
    float alw = 0.f;
    if (lane < 4) alw = alpha[(size_t)widx * 4 + lane] / z[(size_t)dst * 4 + lane];
    float4 hv = *(const float4*)(hbuf + (size_t)src * F_DIM + lane * 4);
    if (lane < 4) alpha[(size_t)widx * 4 + lane] = alw;  // final alpha output
    float* op = acc + (size_t)dst * F_DIM + lane * 4;
    atomicAdd(op + 0, al * hv.x);
    atomicAdd(op + 1, al * hv.y);
    atomicAdd(op + 2, al * hv.z);
    atomicAdd(op + 3, al * hv.w);
}

// ---------------- bias add (x_g_out = acc2 + b2) ----------------
__global__ void add_bias_kernel(const float* __restrict__ acc, const float* __restrict__ b,
                                float* __restrict__ out, long total) {
    long id = (long)blockIdx.x * blockDim.x + threadIdx.x;
    if (id >= total) return;
    out[id] = acc[id] + b[id & (F_DIM - 1)];
}

// ---------------- group attention + final mix ----------------
// One wave per node; lane owns channels [lane*4, lane*4+4).
__global__ void group_kernel(const float* __restrict__ xacc, const float* __restrict__ b1,
                             const float* __restrict__ xg, const int* __restrict__ gidx,
                             const int* __restrict__ nbr, float* __restrict__ xfinal,
                             int Nn) {
    int lane = threadIdx.x & 31;
    int node = blockIdx.x * (blockDim.x >> 5) + (threadIdx.x >> 5);
    if (node >= Nn) return;
    float4 bv = *(const float4*)(b1 + lane * 4);
    float4 xo = *(const float4*)(xacc + (size_t)node * F_DIM + lane * 4);
    xo.x += bv.x; xo.y += bv.y; xo.z += bv.z; xo.w += bv.w;   // x_out = acc + b1
    int gi = gidx[node];
    int idx[17];
    idx[0] = gi;
#pragma unroll
    for (int k = 0; k < 16; ++k) idx[k + 1] = nbr[(size_t)gi * 16 + k];

    float4 cv[17];
    float sv[17];
#pragma unroll
    for (int k = 0; k < 17; ++k) {
        float4 c = *(const float4*)(xg + (size_t)idx[k] * F_DIM + lane * 4);
        cv[k] = c;
        float s = xo.x * c.x + xo.y * c.y + xo.z * c.z + xo.w * c.w;
#pragma unroll
        for (int off = 16; off > 0; off >>= 1) s += __shfl_xor(s, off, 32);
        sv[k] = s;
    }
    float mx = sv[0];
#pragma unroll
    for (int k = 1; k < 17; ++k) mx = fmaxf(mx, sv[k]);
    float sum = 0.f;
#pragma unroll
    for (int k = 0; k < 17; ++k) { sv[k] = __expf(sv[k] - mx); sum += sv[k]; }
    float inv = 1.f / sum;
    float4 go = make_float4(0.f, 0.f, 0.f, 0.f);
#pragma unroll
    for (int k = 0; k < 17; ++k) {
        float w = sv[k] * inv;
        go.x = fmaf(w, cv[k].x, go.x);
        go.y = fmaf(w, cv[k].y, go.y);
        go.z = fmaf(w, cv[k].z, go.z);
        go.w = fmaf(w, cv[k].w, go.w);
    }
    float4 o;
    o.x = 0.5f * (xo.x + go.x);
    o.y = 0.5f * (xo.y + go.y);
    o.z = 0.5f * (xo.z + go.z);
    o.w = 0.5f * (xo.w + go.w);
    *(float4*)(xfinal + (size_t)node * F_DIM + lane * 4) = o;
}

// ============================================================================
extern "C" void kernel_launch(void* const* d_in, const int* in_sizes, int n_in,
                              void* d_out, int out_size, void* d_ws, size_t ws_size,
                              hipStream_t stream) {
    const float* x       = (const float*)d_in[0];
    const int*   ei      = (const int*)d_in[1];
    const float* x_g     = (const float*)d_in[2];
    const int*   ei_g    = (const int*)d_in[3];
    const int*   gidx    = (const int*)d_in[4];
    const int*   nbr     = (const int*)d_in[5];
    const float* W1      = (const float*)d_in[6];
    const float* a_src1  = (const float*)d_in[7];
    const float* a_dst1  = (const float*)d_in[8];
    const float* b1      = (const float*)d_in[9];
    const float* W2      = (const float*)d_in[10];
    const float* a_src2  = (const float*)d_in[11];
    const float* a_dst2  = (const float*)d_in[12];
    const float* b2      = (const float*)d_in[13];

    const int N  = in_sizes[0] / F_DIM;
    const int E  = in_sizes[1] / 2;
    const int K  = in_sizes[2] / F_DIM;
    const int Eg = in_sizes[3] / 2;

    // ---- d_out layout: x_final[N,128] | x_g_out[K,128] | alpha1[E+N,4] | alpha2[Eg+K,4]
    float* out     = (float*)d_out;
    float* x_final = out;
    float* xg_out  = out + (size_t)N * F_DIM;
    float* alpha1  = xg_out + (size_t)K * F_DIM;
    float* alpha2  = alpha1 + (size_t)(E + N) * H_HEADS;

    // ---- workspace layout
    float* ws = (float*)d_ws;
    size_t off = 0;
    float* h1    = ws + off; off += (size_t)N * F_DIM;
    float* h2    = ws + off; off += (size_t)K * F_DIM;
    float* as1   = ws + off; off += (size_t)N * H_HEADS;
    float* ad1   = ws + off; off += (size_t)N * H_HEADS;
    float* m1    = ws + off; off += (size_t)N * H_HEADS;
    float* z1    = ws + off; off += (size_t)N * H_HEADS;
    float* as2   = ws + off; off += (size_t)K * H_HEADS;
    float* ad2   = ws + off; off += (size_t)K * H_HEADS;
    float* m2    = ws + off; off += (size_t)K * H_HEADS;
    float* z2    = ws + off; off += (size_t)K * H_HEADS;
    float* xacc1 = ws + off; off += (size_t)N * F_DIM;
    float* xacc2 = ws + off; off += (size_t)K * F_DIM;

    const int T = 256;
    auto blk = [](long n, int t) { return (int)((n + t - 1) / t); };

    // ---- init accumulators
    fill_kernel<<<1024, T, 0, stream>>>(xacc1, (long)N * F_DIM, 0.f);
    fill_kernel<<<256,  T, 0, stream>>>(xacc2, (long)K * F_DIM, 0.f);
    fill_kernel<<<256,  T, 0, stream>>>(z1, (long)N * H_HEADS, 0.f);
    fill_kernel<<<64,   T, 0, stream>>>(z2, (long)K * H_HEADS, 0.f);
    fill_kernel<<<256,  T, 0, stream>>>(m1, (long)N * H_HEADS, -INFINITY);
    fill_kernel<<<64,   T, 0, stream>>>(m2, (long)K * H_HEADS, -INFINITY);

    // ---- GEMMs (WMMA) with fused attention coefficients; 64 KB dynamic LDS
    gemm_wmma_kernel<<<blk(N, 128), T, F_DIM * F_DIM * sizeof(float), stream>>>(
        x, W1, a_src1, a_dst1, h1, as1, ad1, N);
    gemm_wmma_kernel<<<blk(K, 128), T, F_DIM * F_DIM * sizeof(float), stream>>>(
        x_g, W2, a_src2, a_dst2, h2, as2, ad2, K);

    // ---- edge scores + segment max
    score_kernel<<<blk((long)E + N, T), T, 0, stream>>>(as1, ad1, ei, E, N, alpha1, m1);
    score_kernel<<<blk((long)Eg + K, T), T, 0, stream>>>(as2, ad2, ei_g, Eg, K, alpha2, m2);

    // ---- exp + segment sum
    expsum_kernel<<<blk((long)E + N, T), T, 0, stream>>>(ei, E, N, m1, alpha1, z1);
    expsum_kernel<<<blk((long)Eg + K, T), T, 0, stream>>>(ei_g, Eg, K, m2, alpha2, z2);

    // ---- normalize + aggregate (wave per edge; 8 waves per block)
    agg_kernel<<<blk((long)E + N, 8), T, 0, stream>>>(h1, z1, ei, E, N, alpha1, xacc1);
    agg_kernel<<<blk((long)Eg + K, 8), T, 0, stream>>>(h2, z2, ei_g, Eg, K, alpha2, xacc2);

    // ---- finalize x_g_out = acc2 + b2 (output #2)
    add_bias_kernel<<<blk((long)K * F_DIM, T), T, 0, stream>>>(xacc2, b2, xg_out, (long)K * F_DIM);

    // ---- group attention + final mix (wave per node)
    group_kernel<<<blk(N, 8), T, 0, stream>>>(xacc1, b1, xg_out, gidx, nbr, x_final, N);
}